// _JointODEFunc_47605417509226
// MI455X (gfx1250) — compile-verified
//
#include <hip/hip_runtime.h>
#include <math.h>

// ---------------------------------------------------------------------------
// B=128, N=512, D=1024, C=512
//   out[b, 0:512]    = da = tanh(a@Wa + ctx@Wc + ba)
//   out[b, 512:1536] = dz[b,i] = sum_k c[b,k] * z[b,(i+k)%1024]
// a = s[:, :512], z = s[:, 512:], c = a@Wl + bl staged in d_ws.
// fp32 throughout; GEMMs on V_WMMA_F32_16X16X4_F32 with B panels staged into
// LDS via async global->LDS DMA (ASYNCcnt) in a pair-swizzled layout so each
// B fragment is a single ds_load_b64.
// ---------------------------------------------------------------------------

typedef float v2f __attribute__((ext_vector_type(2)));
typedef float v8f __attribute__((ext_vector_type(8)));

typedef __attribute__((address_space(1))) int g_i32;   // global int
typedef __attribute__((address_space(3))) int l_i32;   // LDS int

#if __has_builtin(__builtin_amdgcn_global_load_async_to_lds_b32)
#define HAVE_ASYNC_LDS 1
#else
#define HAVE_ASYNC_LDS 0
#endif

__device__ __forceinline__ void wait_async0() {
#if __has_builtin(__builtin_amdgcn_s_wait_asynccnt)
    __builtin_amdgcn_s_wait_asynccnt(0);
#else
    asm volatile("s_wait_asynccnt 0" ::: "memory");
#endif
}

// D = A(16x4) * B(4x16) + C, fp32, wave32.
__device__ __forceinline__ v8f wmma_f32(v2f a, v2f b, v8f c) {
    return __builtin_amdgcn_wmma_f32_16x16x4_f32(
        /*neg_a=*/false, a, /*neg_b=*/false, b,
        /*c_mod=*/(short)0, c, /*reuse_a=*/false, /*reuse_b=*/false);
}

#define KC 64                      // K rows of B staged per chunk
#define BBUF_FLOATS (KC * 32)      // 8 KB: pair-swizzled [k/2][n][k%2]

// Accumulate acc += A[:, kbase:kbase+Ktot] @ W[kbase.., n0:n0+32] for this
// wave's 16x32 tile. Block = 256 threads (8 waves, one M-tile each); B chunk
// staged cooperatively into LDS once per block and read as ds_load_b64.
__device__ __forceinline__ void gemm_accum(
    const float* __restrict__ arow,   // per-lane A row base (row mrow, col 0)
    const float* __restrict__ W, int ldb, int Ktot, int n0,
    float* bbuf, int tid, int lane, v8f& acc0, v8f& acc1)
{
    const int khalf = (lane >> 4) << 1;   // lanes 0-15 -> K+0/1, 16-31 -> K+2/3
    const int ncl   = (lane & 15) << 1;   // pair-swizzled column offset

    for (int kc = 0; kc < Ktot; kc += KC) {
        __syncthreads();                  // previous chunk fully consumed
        // ---- stage B chunk: W[kc+k, n0+n] -> bbuf[(k/2)*64 + n*2 + (k&1)]
        #pragma unroll
        for (int j = 0; j < BBUF_FLOATS / 256; ++j) {
            const int e = tid + 256 * j;
            const int k = e >> 5;
            const int n = e & 31;
            const int f = ((k >> 1) << 6) + (n << 1) + (k & 1);
            const float* gp = W + (kc + k) * ldb + n0 + n;
#if HAVE_ASYNC_LDS
            __builtin_amdgcn_global_load_async_to_lds_b32(
                (g_i32*)gp, (l_i32*)&bbuf[f], 0, 0);
#else
            bbuf[f] = *gp;
#endif
        }
#if HAVE_ASYNC_LDS
        wait_async0();
#endif
        __syncthreads();

        // ---- compute 16 k-steps on the staged chunk
        #pragma unroll 4
        for (int kl = 0; kl < KC; kl += 4) {
            const int kp = (kl + khalf) >> 1;       // pair row in chunk
            v2f af = *(const v2f*)(arow + kc + kl + khalf);       // global b64
            v2f b0 = *(const v2f*)(bbuf + (kp << 6) + ncl);       // ds b64
            v2f b1 = *(const v2f*)(bbuf + (kp << 6) + ncl + 32);  // ds b64
            acc0 = wmma_f32(af, b0, acc0);
            acc1 = wmma_f32(af, b1, acc1);
        }
    }
}

// ---------------------------------------------------------------------------
// Kernel 1: da = tanh(a @ Wa + ctx @ Wc + ba)  (M=128, K=512+512, Nn=512)
// grid.x = 16 N-strips of 32; block = 8 waves, wave w owns rows 16w..16w+15.
// ---------------------------------------------------------------------------
__global__ __launch_bounds__(256)
void da_wmma_kernel(const float* __restrict__ s,
                    const float* __restrict__ ctx,
                    const float* __restrict__ Wa,
                    const float* __restrict__ Wc,
                    const float* __restrict__ ba,
                    float* __restrict__ out)
{
    __shared__ float bbuf[BBUF_FLOATS];
    const int tid  = threadIdx.x;
    const int lane = tid & 31;
    const int m0   = (tid >> 5) << 4;     // wave id * 16
    const int n0   = blockIdx.x << 5;
    const int mrow = m0 + (lane & 15);
    const int nc0  = n0 + (lane & 15);
    const int nc1  = nc0 + 16;

    v8f acc0 = {}; v8f acc1 = {};
    gemm_accum(s   + mrow * 1536, Wa, 512, 512, n0, bbuf, tid, lane, acc0, acc1);
    gemm_accum(ctx + mrow * 512,  Wc, 512, 512, n0, bbuf, tid, lane, acc0, acc1);

    const float bias0 = ba[nc0];
    const float bias1 = ba[nc1];
    #pragma unroll
    for (int v = 0; v < 8; ++v) {
        const int row = m0 + v + ((lane >> 4) << 3);   // C/D VGPR layout
        out[row * 1536 + nc0] = tanhf(acc0[v] + bias0);
        out[row * 1536 + nc1] = tanhf(acc1[v] + bias1);
    }
}

// ---------------------------------------------------------------------------
// Kernel 2: c = a @ Wl + bl  (M=128, K=512, Nn=1024) -> d_ws (row stride 1024)
// grid.x = 32 N-strips of 32.
// ---------------------------------------------------------------------------
__global__ __launch_bounds__(256)
void c_wmma_kernel(const float* __restrict__ s,
                   const float* __restrict__ Wl,
                   const float* __restrict__ bl,
                   float* __restrict__ c)
{
    __shared__ float bbuf[BBUF_FLOATS];
    const int tid  = threadIdx.x;
    const int lane = tid & 31;
    const int m0   = (tid >> 5) << 4;
    const int n0   = blockIdx.x << 5;
    const int mrow = m0 + (lane & 15);
    const int nc0  = n0 + (lane & 15);
    const int nc1  = nc0 + 16;

    v8f acc0 = {}; v8f acc1 = {};
    gemm_accum(s + mrow * 1536, Wl, 1024, 512, n0, bbuf, tid, lane, acc0, acc1);

    const float bias0 = bl[nc0];
    const float bias1 = bl[nc1];
    #pragma unroll
    for (int v = 0; v < 8; ++v) {
        const int row = m0 + v + ((lane >> 4) << 3);
        c[row * 1024 + nc0] = acc0[v] + bias0;
        c[row * 1024 + nc1] = acc1[v] + bias1;
    }
}

// ---------------------------------------------------------------------------
// Kernel 3: dz[b,i] = sum_k c[b,k] * z[b,(i+k)%1024]
// One block per batch, 256 threads; c[b] (4KB) + doubled z[b] (8KB) in LDS.
// cs[k] is a lane-uniform broadcast; zz[tid+off+k] is stride-1 across lanes
// -> conflict-free. Doubled z removes the modulo (max idx 2046 < 2048).
// ---------------------------------------------------------------------------
__global__ __launch_bounds__(256)
void dz_kernel(const float* __restrict__ s,
               const float* __restrict__ c,
               float* __restrict__ out)
{
    __shared__ float cs[1024];
    __shared__ float zz[2048];

    const int b   = blockIdx.x;
    const int tid = threadIdx.x;

    const float* crow = c + b * 1024;
    const float* zrow = s + b * 1536 + 512;   // z = s[:, 512:]

    #pragma unroll
    for (int i = tid; i < 1024; i += 256) {
        const float zv = zrow[i];
        cs[i]        = crow[i];
        zz[i]        = zv;
        zz[i + 1024] = zv;
    }
    __syncthreads();

    float acc0 = 0.f, acc1 = 0.f, acc2 = 0.f, acc3 = 0.f;
    #pragma unroll 8
    for (int k = 0; k < 1024; ++k) {
        const float ck = cs[k];
        acc0 = fmaf(ck, zz[tid       + k], acc0);
        acc1 = fmaf(ck, zz[tid + 256 + k], acc1);
        acc2 = fmaf(ck, zz[tid + 512 + k], acc2);
        acc3 = fmaf(ck, zz[tid + 768 + k], acc3);
    }

    float* orow = out + b * 1536 + 512;
    orow[tid      ] = acc0;
    orow[tid + 256] = acc1;
    orow[tid + 512] = acc2;
    orow[tid + 768] = acc3;
}

// ---------------------------------------------------------------------------
// Launch. Inputs (setup_inputs order): t, s, context, Wa, Wc, ba, Wl, bl
// ---------------------------------------------------------------------------
extern "C" void kernel_launch(void* const* d_in, const int* in_sizes, int n_in,
                              void* d_out, int out_size, void* d_ws, size_t ws_size,
                              hipStream_t stream) {
    const float* s   = (const float*)d_in[1];
    const float* ctx = (const float*)d_in[2];
    const float* Wa  = (const float*)d_in[3];
    const float* Wc  = (const float*)d_in[4];
    const float* ba  = (const float*)d_in[5];
    const float* bl  = (const float*)d_in[7];
    const float* Wl  = (const float*)d_in[6];
    float* out  = (float*)d_out;
    float* c_ws = (float*)d_ws;               // 128*1024 floats = 512 KB

    da_wmma_kernel<<<dim3(16), 256, 0, stream>>>(s, ctx, Wa, Wc, ba, out);
    c_wmma_kernel<<<dim3(32), 256, 0, stream>>>(s, Wl, bl, c_ws);
    dz_kernel<<<dim3(128), 256, 0, stream>>>(s, c_ws, out);   // depends on c_ws
}